// TriplaneEncoding_78202764525718
// MI455X (gfx1250) — compile-verified
//
#include <hip/hip_runtime.h>
#include <hip/hip_bf16.h>
#include <stdint.h>

#define RES 512
#define NCH 64

typedef float v4f __attribute__((ext_vector_type(4)));
typedef float v2f __attribute__((ext_vector_type(2)));

struct PlaneSample {
    int x0, x1, y0, y1;          // clamped texel coords
    float w00, w01, w10, w11;    // bilinear weights * validity masks
};

__device__ __forceinline__ PlaneSample plane_setup(float u, float v) {
    const float s = 0.5f * (float)(RES - 1);
    float gx = (u + 1.0f) * s;
    float gy = (v + 1.0f) * s;
    float fx = floorf(gx), fy = floorf(gy);
    float wx = gx - fx,  wy = gy - fy;
    int x0 = (int)fx, y0 = (int)fy;
    int x1 = x0 + 1,  y1 = y0 + 1;
    float vx0 = (x0 >= 0 && x0 < RES) ? 1.0f : 0.0f;
    float vx1 = (x1 >= 0 && x1 < RES) ? 1.0f : 0.0f;
    float vy0 = (y0 >= 0 && y0 < RES) ? 1.0f : 0.0f;
    float vy1 = (y1 >= 0 && y1 < RES) ? 1.0f : 0.0f;
    PlaneSample r;
    r.x0 = min(max(x0, 0), RES - 1);
    r.x1 = min(max(x1, 0), RES - 1);
    r.y0 = min(max(y0, 0), RES - 1);
    r.y1 = min(max(y1, 0), RES - 1);
    r.w00 = (1.0f - wx) * (1.0f - wy) * vx0 * vy0;
    r.w01 = wx * (1.0f - wy) * vx1 * vy0;
    r.w10 = (1.0f - wx) * wy * vx0 * vy1;
    r.w11 = wx * wy * vx1 * vy1;
    return r;
}

// ---------------------------------------------------------------------------
// Pass 1: transpose (3, C, R, R) -> (3, R, R, C) so one texel's 64 channels
// are a contiguous 256B block. All global traffic as B128; source reads are
// non-temporal (single-use) so the transposed copy owns the 192MB L2.
// ---------------------------------------------------------------------------
__global__ __launch_bounds__(256)
void transpose_planes(const float* __restrict__ src, float* __restrict__ dst) {
    __shared__ float tile[64][65];              // pad 65: conflict-free col reads
    const int b   = blockIdx.x;                 // (p, y, x_tile)
    const int xt  = b & 7;
    const int y   = (b >> 3) & (RES - 1);
    const int p   = b >> 12;
    const int x0  = xt * 64;
    const int xq  = threadIdx.x & 15;           // 16B chunk (4 floats) along x / c
    const int row = threadIdx.x >> 4;           // 0..15

    #pragma unroll
    for (int c = row; c < NCH; c += 16) {       // coalesced B128 NT src reads
        const v4f* sp = reinterpret_cast<const v4f*>(
            src + (((size_t)p * NCH + c) * RES + y) * RES + x0) + xq;
        v4f v = __builtin_nontemporal_load(sp);
        tile[c][xq * 4 + 0] = v.x;
        tile[c][xq * 4 + 1] = v.y;
        tile[c][xq * 4 + 2] = v.z;
        tile[c][xq * 4 + 3] = v.w;
    }
    __syncthreads();
    #pragma unroll
    for (int xx = row; xx < 64; xx += 16) {     // coalesced B128 dst writes along c
        v4f v;
        v.x = tile[xq * 4 + 0][xx];
        v.y = tile[xq * 4 + 1][xx];
        v.z = tile[xq * 4 + 2][xx];
        v.w = tile[xq * 4 + 3][xx];
        reinterpret_cast<v4f*>(
            dst + (((size_t)p * RES + y) * RES + (x0 + xx)) * NCH)[xq] = v;
    }
}

// ---------------------------------------------------------------------------
// Pass 2: one wave per point. 12 corner blocks (3 planes x 4 corners x 256B)
// staged into LDS with async gather-to-LDS (ASYNCcnt path), then combined
// with bilinear weights. Lane l handles channels 2l, 2l+1. Output stores are
// non-temporal so the 256MB result doesn't evict planesT from L2.
// ---------------------------------------------------------------------------
__global__ __launch_bounds__(256)
void triplane_gather_async(const float* __restrict__ pts,
                           const float* __restrict__ planesT,
                           float* __restrict__ out, int npts) {
    __shared__ float smem[8 * 12 * NCH];        // 8 waves * 12 corners * 256B = 24KB
    const int lane = threadIdx.x & 31;
    const int wave = threadIdx.x >> 5;
    const int n    = blockIdx.x * 8 + wave;
    if (n >= npts) return;                      // whole-wave uniform

    const float p0 = pts[3 * (size_t)n + 0];
    const float p1 = pts[3 * (size_t)n + 1];
    const float p2 = pts[3 * (size_t)n + 2];

    PlaneSample s[3];
    s[0] = plane_setup(p0, p1);                 // plane 0: (x, y)
    s[1] = plane_setup(p0, p2);                 // plane 1: (x, z)
    s[2] = plane_setup(p1, p2);                 // plane 2: (y, z)

    int   bi[12];
    float wg[12];
    #pragma unroll
    for (int p = 0; p < 3; ++p) {
        const int pb = p * RES * RES;
        bi[p*4+0] = (pb + s[p].y0 * RES + s[p].x0) * NCH;  wg[p*4+0] = s[p].w00;
        bi[p*4+1] = (pb + s[p].y0 * RES + s[p].x1) * NCH;  wg[p*4+1] = s[p].w01;
        bi[p*4+2] = (pb + s[p].y1 * RES + s[p].x0) * NCH;  wg[p*4+2] = s[p].w10;
        bi[p*4+3] = (pb + s[p].y1 * RES + s[p].x1) * NCH;  wg[p*4+3] = s[p].w11;
    }

    // LDS byte offset of this wave's staging region (flat addr low 32b = LDS off)
    const uint32_t ldsWave = (uint32_t)(uintptr_t)(&smem[wave * 12 * NCH]);
    const uint32_t t16 = (uint32_t)(lane & 15) * 16u;   // 16B chunk within corner
    const uint32_t hi  = (lane & 16) ? 256u : 0u;       // lanes 16..31 -> 2nd corner

    #pragma unroll
    for (int j = 0; j < 6; ++j) {               // each op moves 2 corners (512B)
        const int      b = (lane & 16) ? bi[2*j + 1] : bi[2*j];
        const char*    g = (const char*)planesT + (size_t)b * 4 + t16;
        const uint32_t l = ldsWave + (uint32_t)(2 * j) * 256u + hi + t16;
        asm volatile("global_load_async_to_lds_b128 %0, %1, off"
                     :: "v"(l), "v"(g) : "memory");
    }
    asm volatile("s_wait_asynccnt 0x0" ::: "memory");

    const float* sm = &smem[wave * 12 * NCH];
    const int c = lane * 2;
    float a0 = 0.0f, a1 = 0.0f;
    #pragma unroll
    for (int k = 0; k < 12; ++k) {              // ds_load_b64, bank-conflict-free
        a0 += wg[k] * sm[k * NCH + c];
        a1 += wg[k] * sm[k * NCH + c + 1];
    }
    v2f o; o.x = a0; o.y = a1;
    __builtin_nontemporal_store(o, reinterpret_cast<v2f*>(out + (size_t)n * NCH + c));
}

// ---------------------------------------------------------------------------
// Fallback (ws too small): gather straight from (3, C, R, R), thread-per-channel.
// ---------------------------------------------------------------------------
__global__ __launch_bounds__(256)
void triplane_gather_direct(const float* __restrict__ pts,
                            const float* __restrict__ plane,
                            float* __restrict__ out, int npts) {
    const int n = blockIdx.x * 4 + (int)(threadIdx.x >> 6);
    const int c = threadIdx.x & 63;
    if (n >= npts) return;

    const float p0 = pts[3 * (size_t)n + 0];
    const float p1 = pts[3 * (size_t)n + 1];
    const float p2 = pts[3 * (size_t)n + 2];

    float acc = 0.0f;
    #pragma unroll
    for (int p = 0; p < 3; ++p) {
        PlaneSample sp = (p == 0) ? plane_setup(p0, p1)
                       : (p == 1) ? plane_setup(p0, p2)
                                  : plane_setup(p1, p2);
        const float* f = plane + (((size_t)p * NCH + c) * RES) * RES;
        acc += sp.w00 * f[(size_t)sp.y0 * RES + sp.x0];
        acc += sp.w01 * f[(size_t)sp.y0 * RES + sp.x1];
        acc += sp.w10 * f[(size_t)sp.y1 * RES + sp.x0];
        acc += sp.w11 * f[(size_t)sp.y1 * RES + sp.x1];
    }
    out[(size_t)n * NCH + c] = acc;
}

extern "C" void kernel_launch(void* const* d_in, const int* in_sizes, int n_in,
                              void* d_out, int out_size, void* d_ws, size_t ws_size,
                              hipStream_t stream) {
    const float* pts   = (const float*)d_in[0];   // (N, 3) fp32
    const float* plane = (const float*)d_in[1];   // (3, 64, 512, 512) fp32
    float*       out   = (float*)d_out;           // (N, 64) fp32
    const int npts = in_sizes[0] / 3;

    const size_t needed = (size_t)3 * RES * RES * NCH * sizeof(float); // 192 MiB
    if (ws_size >= needed && d_ws != nullptr) {
        float* planesT = (float*)d_ws;
        transpose_planes<<<3 * RES * 8, 256, 0, stream>>>(plane, planesT);
        triplane_gather_async<<<(npts + 7) / 8, 256, 0, stream>>>(pts, planesT, out, npts);
    } else {
        triplane_gather_direct<<<(npts + 3) / 4, 256, 0, stream>>>(pts, plane, out, npts);
    }
}